// AsymmetricSVD_84361747628057
// MI455X (gfx1250) — compile-verified
//
#include <hip/hip_runtime.h>

#define FDIM 128
#define MU_C 3.5f

typedef __attribute__((ext_vector_type(2))) float v2f;
typedef __attribute__((ext_vector_type(8))) float v8f;

// ---------------- Kernel 1: segment boundaries via binary search -------------
// segment_ids is sorted; start[b] = lower_bound(seg, b), b in [0, B].
__global__ void asvd_seg_bounds(const int* __restrict__ seg, int T, int B,
                                int* __restrict__ start) {
    int b = blockIdx.x * blockDim.x + threadIdx.x;
    if (b > B) return;
    int lo = 0, hi = T;
    while (lo < hi) {
        int mid = (lo + hi) >> 1;
        if (seg[mid] < b) lo = mid + 1; else hi = mid;
    }
    start[b] = lo;
}

// ---------------- Kernel 2: ragged weighted gather-sum -----------------------
// One wave32 per query b. Lane l covers features [4l, 4l+4) as a float4.
// acc += w_t * X[j_t] + Y[j_t]; then Z[b] = norm * acc.
__global__ void asvd_accumulate(const float* __restrict__ bu,
                                const float* __restrict__ bi,
                                const float* __restrict__ X,
                                const float* __restrict__ Y,
                                const int* __restrict__ user,
                                const int* __restrict__ imp_items,
                                const int* __restrict__ imp_ratings,
                                const int* __restrict__ start,
                                float* __restrict__ Z,
                                int B) {
    const int wid  = threadIdx.x >> 5;
    const int lane = threadIdx.x & 31;
    const int b = blockIdx.x * (blockDim.x >> 5) + wid;
    if (b >= B) return;

    const int s = start[b];
    const int e = start[b + 1];
    const float bu_u = bu[user[b]];           // uniform across the wave

    const float4* __restrict__ X4 = reinterpret_cast<const float4*>(X);
    const float4* __restrict__ Y4 = reinterpret_cast<const float4*>(Y);

    float4 acc = make_float4(0.f, 0.f, 0.f, 0.f);
    for (int t = s; t < e; ++t) {
        const int   j = imp_items[t];                       // uniform -> s_load
        const float w = (float)imp_ratings[t] - (MU_C + bu_u + bi[j]);
        const float4 xv = X4[(size_t)j * 32 + lane];        // 512B/row per wave
        const float4 yv = Y4[(size_t)j * 32 + lane];
        if (t + 1 < e) {                                    // hide gather latency
            const int jn = imp_items[t + 1];
            __builtin_prefetch(&X4[(size_t)jn * 32 + lane], 0, 3);
            __builtin_prefetch(&Y4[(size_t)jn * 32 + lane], 0, 3);
        }
        acc.x = fmaf(w, xv.x, acc.x) + yv.x;
        acc.y = fmaf(w, xv.y, acc.y) + yv.y;
        acc.z = fmaf(w, xv.z, acc.z) + yv.z;
        acc.w = fmaf(w, xv.w, acc.w) + yv.w;
    }

    const int cnt = e - s;
    const float norm = (cnt > 0) ? rsqrtf((float)cnt) : 1.0f;
    float4 z;
    z.x = norm * acc.x; z.y = norm * acc.y;
    z.z = norm * acc.z; z.w = norm * acc.w;
    reinterpret_cast<float4*>(Z)[(size_t)b * 32 + lane] = z;
}

// ---------------- Kernel 3: batched dot via V_WMMA_F32_16X16X4_F32 -----------
// One wave per 16 queries. A = 16 gathered Q rows (16xK), B fragments hold
// z^T (Kx16), accumulated over K=128 in steps of 4. D's diagonal = the dots.
// A-layout (32b, 16x4): lanes 0-15 hold K=k0,k0+1; lanes 16-31 hold k0+2,k0+3.
// B-layout (32b, 4x16): same striping with N = lane&15. Both fragments are the
// float2 at [row*128 + k0 + 2*(lane>>4)].
__global__ void asvd_wmma_dot(const float* __restrict__ Q,
                              const float* __restrict__ Z,
                              const float* __restrict__ bu,
                              const float* __restrict__ bi,
                              const int* __restrict__ user,
                              const int* __restrict__ item,
                              float* __restrict__ out) {
    const int base  = blockIdx.x * 16;
    const int L     = threadIdx.x;        // 0..31, wave32, no divergence
    const int row   = L & 15;
    const int koff  = (L >> 4) << 1;      // 0 or 2
    const int gq    = base + row;

    const float* __restrict__ qrow = Q + (size_t)item[gq] * FDIM;
    const float* __restrict__ zrow = Z + (size_t)gq * FDIM;

    v8f c = {};
    #pragma unroll
    for (int k0 = 0; k0 < FDIM; k0 += 4) {
        const v2f a  = *reinterpret_cast<const v2f*>(qrow + k0 + koff);
        const v2f bm = *reinterpret_cast<const v2f*>(zrow + k0 + koff);
        c = __builtin_amdgcn_wmma_f32_16x16x4_f32(
                /*neg_a=*/false, a, /*neg_b=*/false, bm,
                /*c_mod=*/(short)0, c, /*reuse_a=*/false, /*reuse_b=*/false);
    }

    // Diagonal extraction per C/D layout: (m,m) for m=0..7 lives in lane m,
    // VGPR m; for m=8..15 in lane m+16, VGPR m-8.
    const bool active = (L < 8) || (L >= 16 && L < 24);
    const int  ci     = (L < 8) ? L : ((L >= 16 && L < 24) ? (L - 16) : 0);
    const int  qloc   = (L < 8) ? L : (L - 8);
    float d = 0.f;
    switch (ci) {
        case 0: d = c[0]; break; case 1: d = c[1]; break;
        case 2: d = c[2]; break; case 3: d = c[3]; break;
        case 4: d = c[4]; break; case 5: d = c[5]; break;
        case 6: d = c[6]; break; case 7: d = c[7]; break;
    }
    if (active) {
        const int q = base + qloc;
        const float bui = MU_C + bu[user[q]] + bi[item[q]];
        out[q] = bui + d;
    }
}

// ---------------- Tail (B % 16 != 0; not hit for B = 4096) -------------------
__global__ void asvd_tail_dot(const float* __restrict__ Q,
                              const float* __restrict__ Z,
                              const float* __restrict__ bu,
                              const float* __restrict__ bi,
                              const int* __restrict__ user,
                              const int* __restrict__ item,
                              float* __restrict__ out, int q0, int B) {
    int q = q0 + blockIdx.x * blockDim.x + threadIdx.x;
    if (q >= B) return;
    const float* qr = Q + (size_t)item[q] * FDIM;
    const float* zr = Z + (size_t)q * FDIM;
    float d = 0.f;
    for (int k = 0; k < FDIM; ++k) d = fmaf(qr[k], zr[k], d);
    out[q] = MU_C + bu[user[q]] + bi[item[q]] + d;
}

extern "C" void kernel_launch(void* const* d_in, const int* in_sizes, int n_in,
                              void* d_out, int out_size, void* d_ws, size_t ws_size,
                              hipStream_t stream) {
    const float* bu          = (const float*)d_in[0];
    const float* bi          = (const float*)d_in[1];
    const float* Q           = (const float*)d_in[2];
    const float* X           = (const float*)d_in[3];
    const float* Y           = (const float*)d_in[4];
    const int*   user        = (const int*)d_in[5];
    const int*   item        = (const int*)d_in[6];
    const int*   imp_items   = (const int*)d_in[7];
    const int*   imp_ratings = (const int*)d_in[8];
    const int*   segment_ids = (const int*)d_in[9];
    float* out = (float*)d_out;

    const int B = in_sizes[5];
    const int T = in_sizes[7];

    // Workspace layout: Z [B*F floats], then seg_start [B+1 ints].
    float* Z         = (float*)d_ws;
    int*   seg_start = (int*)(Z + (size_t)B * FDIM);

    // 1) segment boundaries
    {
        int n = B + 1;
        asvd_seg_bounds<<<(n + 255) / 256, 256, 0, stream>>>(segment_ids, T, B, seg_start);
    }
    // 2) ragged weighted gather-sum: 8 waves (8 queries) per 256-thread block
    {
        int qpb = 8;
        asvd_accumulate<<<(B + qpb - 1) / qpb, 32 * qpb, 0, stream>>>(
            bu, bi, X, Y, user, imp_items, imp_ratings, seg_start, Z, B);
    }
    // 3) WMMA epilogue: one wave per 16 queries
    {
        int nblk = B / 16;
        if (nblk > 0) {
            asvd_wmma_dot<<<nblk, 32, 0, stream>>>(Q, Z, bu, bi, user, item, out);
        }
        int q0 = nblk * 16;
        int tail = B - q0;
        if (tail > 0) {
            asvd_tail_dot<<<(tail + 63) / 64, 64, 0, stream>>>(Q, Z, bu, bi, user, item, out, q0, B);
        }
    }
}